// RCNNProposal_52639119180043
// MI455X (gfx1250) — compile-verified
//
#include <hip/hip_runtime.h>
#include <hip/hip_bf16.h>
#include <cstdint>

// ---------------- problem constants (match reference) ----------------
#define NCLS      5          // C + 1
#define MAX_CAND  300
#define TOP_N     100
#define OVERLAP   0.5f
#define THR_OBJ   0.2f
#define THR_CLS   0.1f
#define MIN_WH    6.0f

// ---------------- selection machinery sizes ----------------
#define HIST_BINS 65536
#define CAPA      512        // entries with key16 > T   (provably < 300)
#define CAPB      3584       // entries with key16 == T  (bin occupancy cap)
#define SORTN     4096       // CAPA + CAPB, power of two for bitonic sort
#define NMS_THREADS 1024
#define KEEPW     ((MAX_CAND + 31) / 32)    // 10 words of keep/adjacency bits

// ws layout in uint32 units (list region byte offset 65544*4=262176, 8B aligned)
#define WS_HIST   0
#define WS_CTRS   (WS_HIST + HIST_BINS)          // [0]=cntA [1]=cntB [2]=T
#define WS_LIST   (WS_CTRS + 8)                  // SORTN u64 composites
#define WS_KEYS   (WS_LIST + 2 * SORTN)          // n u32 score-bit keys (4 MB)
#define WS_ZERO_U32 WS_KEYS                      // keys are fully overwritten each pass

#define CTR_A 0
#define CTR_B 1
#define CTR_T 2

typedef unsigned int v4u __attribute__((ext_vector_type(4)));
typedef int          v8i __attribute__((ext_vector_type(8)));
typedef int          v4i __attribute__((ext_vector_type(4)));

// Decode one ROI exactly as the reference does; returns validity, score, box.
__device__ __forceinline__ bool decode_one(
    int i, const float* __restrict__ sm, const float* __restrict__ bp,
    const float* __restrict__ rois, float w_img, float h_img,
    float& score, float box[4])
{
  const float* s = sm + (size_t)i * NCLS;
  float p0 = s[0];
  float best = s[1]; int bc = 1;
#pragma unroll
  for (int c = 2; c < NCLS; ++c) { float v = s[c]; if (v > best) { best = v; bc = c; } }

  const float* r = rois + (size_t)i * 5;
  float x1 = r[1], y1 = r[2], x2 = r[3], y2 = r[4];
  float w  = x2 - x1 + 1.0f, h = y2 - y1 + 1.0f;
  float cx = x1 + 0.5f * (w - 1.0f), cy = y1 + 0.5f * (h - 1.0f);

  // 16B-aligned vector load: byte offset 80*i + 16*bc is always 16-aligned
  const float4 dv = *reinterpret_cast<const float4*>(bp + (size_t)i * (4 * NCLS) + 4 * bc);
  float d0 = dv.x * 0.1f, d1 = dv.y * 0.1f, d2 = dv.z * 0.2f, d3 = dv.w * 0.2f;
  float pcx = d0 * w + cx, pcy = d1 * h + cy;
  float pw  = expf(d2) * w, ph = expf(d3) * h;

  float bx0 = pcx - 0.5f * (pw - 1.0f), by0 = pcy - 0.5f * (ph - 1.0f);
  float bx1 = pcx + 0.5f * (pw - 1.0f), by1 = pcy + 0.5f * (ph - 1.0f);
  bx0 = fminf(fmaxf(bx0, 0.0f), w_img - 1.0f);
  by0 = fminf(fmaxf(by0, 0.0f), h_img - 1.0f);
  bx1 = fminf(fmaxf(bx1, 0.0f), w_img - 1.0f);
  by1 = fminf(fmaxf(by1, 0.0f), h_img - 1.0f);
  box[0] = bx0; box[1] = by0; box[2] = bx1; box[3] = by1;

  float bw = bx1 - bx0 + 1.0f, bh = by1 - by0 + 1.0f;
  score = best;
  return ((1.0f - p0) >= THR_OBJ) && (best > THR_CLS) &&
         ((bw >= MIN_WH) || (bh >= MIN_WH));
}

__global__ void k_zero(uint32_t* __restrict__ ws, float* __restrict__ out) {
  int i = blockIdx.x * blockDim.x + threadIdx.x;
  if (i < WS_ZERO_U32) ws[i] = 0u;
  if (i < TOP_N * 10)  out[i] = 0.0f;
}

// Pass 1: full 120 MB stream. Decode, histogram top-16 key bits, cache score bits.
__global__ void k_hist(const float* __restrict__ sm, const float* __restrict__ bp,
                       const float* __restrict__ rois, const float* __restrict__ im_info,
                       uint32_t* __restrict__ ws, int n) {
  int i = blockIdx.x * blockDim.x + threadIdx.x;
  if (i >= n) return;
  float h_img = im_info[0], w_img = im_info[1];
  float score, box[4];
  uint32_t key = 0u;
  if (decode_one(i, sm, bp, rois, w_img, h_img, score, box)) {
    key = __float_as_uint(score);                 // > 0 since score > 0.1
    atomicAdd(&ws[WS_HIST + (key >> 16)], 1u);
  }
  ws[WS_KEYS + i] = key;                          // 0 marks invalid
}

__global__ __launch_bounds__(1024) void k_select(uint32_t* __restrict__ ws) {
  __shared__ uint32_t csum[1024];
  int t = threadIdx.x;
  const int CHUNK = HIST_BINS / 1024;
  uint32_t s = 0;
  int base = t * CHUNK;
  for (int b = 0; b < CHUNK; ++b) s += ws[WS_HIST + base + b];
  csum[t] = s;
  __syncthreads();
  if (t == 0) {
    uint32_t cum = 0, T = 0;
    for (int c = 1023; c >= 0; --c) {
      if (cum + csum[c] >= MAX_CAND) {
        int bb = c * CHUNK;
        for (int b = CHUNK - 1; b >= 0; --b) {
          cum += ws[WS_HIST + bb + b];
          if (cum >= MAX_CAND) { T = (uint32_t)(bb + b); break; }
        }
        break;
      }
      cum += csum[c];
    }
    ws[WS_CTRS + CTR_T] = T;   // if total valid < 300, T stays 0 -> everything goes to A
  }
}

// Pass 2: light 4 MB stream over cached keys only (L2-resident on 192 MB L2).
__global__ void k_compact(uint32_t* __restrict__ ws, int n) {
  int i = blockIdx.x * blockDim.x + threadIdx.x;
  if (i >= n) return;
  uint32_t sb = ws[WS_KEYS + i];
  if (sb == 0u) return;
  uint32_t kb = sb >> 16;
  uint32_t T  = ws[WS_CTRS + CTR_T];
  unsigned long long* list = (unsigned long long*)(ws + WS_LIST);
  // composite: descending score, ascending index tie-break (matches top_k)
  unsigned long long comp =
      ((unsigned long long)sb << 32) | (unsigned long long)(0xFFFFFFFFu - (uint32_t)i);
  if (kb > T) {
    uint32_t p = atomicAdd(&ws[WS_CTRS + CTR_A], 1u);
    if (p < CAPA) list[p] = comp;
  } else if (kb == T) {
    uint32_t p = atomicAdd(&ws[WS_CTRS + CTR_B], 1u);
    if (p < CAPB) list[CAPA + p] = comp;
  }
}

__global__ __launch_bounds__(NMS_THREADS) void k_sort_nms(
    const float* __restrict__ sm, const float* __restrict__ bp,
    const float* __restrict__ rois, const float* __restrict__ im_info,
    const uint32_t* __restrict__ ws, float* __restrict__ out) {
  __shared__ unsigned long long sh_sort[SORTN];     // 32 KB, TDM destination
  __shared__ float    sh_box[MAX_CAND][4];
  __shared__ int      sh_idx[MAX_CAND];
  __shared__ int      sh_keep[MAX_CAND];
  __shared__ int      sh_ord[MAX_CAND];
  __shared__ uint32_t sh_adj[MAX_CAND][KEEPW];      // IoU>thr adjacency bits (j>i only)

  int t = threadIdx.x;
  float h_img = im_info[0], w_img = im_info[1];

  // ---- CDNA5 Tensor Data Mover: DMA candidate lists (global ws) -> LDS ----
  if (t < 32) {   // single wave issues the descriptor (TDM ignores EXEC)
    uint64_t ga  = (uint64_t)(uintptr_t)(ws + WS_LIST);
    uint32_t lds = (uint32_t)(uintptr_t)(void*)&sh_sort[0];   // addr[31:0] = LDS offset
    v4u g0;
    g0.x = 1u;                                       // count=1, user mode
    g0.y = lds;                                      // lds_addr
    g0.z = (uint32_t)(ga & 0xFFFFFFFFull);           // global_addr[31:0]
    g0.w = (uint32_t)((ga >> 32) & 0x1FFFFFFull)     // global_addr[56:32]
         | (2u << 30);                               // type = 2 ("image")
    v8i g1;
    g1[0] = (int)(3u << 16);                         // data_size = 8 bytes
    g1[1] = (int)((SORTN & 0xFFFF) << 16);           // tensor_dim0[15:0]
    g1[2] = (int)((SORTN >> 16) | (1u << 16));       // tensor_dim0[31:16] | tensor_dim1=1
    g1[3] = (int)((SORTN & 0xFFFF) << 16);           // tile_dim0 = SORTN
    g1[4] = 1;                                       // tile_dim1 = 1, tile_dim2 = 0
    g1[5] = SORTN;                                   // tensor_dim0_stride[31:0]
    g1[6] = 0;
    g1[7] = 0;
    v4i gz = {0, 0, 0, 0};
#if defined(__clang_major__) && (__clang_major__ >= 23)
    v8i gz8 = {0, 0, 0, 0, 0, 0, 0, 0};
    __builtin_amdgcn_tensor_load_to_lds(g0, g1, gz, gz, gz8, 0);
#else
    __builtin_amdgcn_tensor_load_to_lds(g0, g1, gz, gz, 0);
#endif
    __builtin_amdgcn_s_wait_tensorcnt(0);
  }
  __syncthreads();

  // ---- bitonic sort SORTN u64 composites, descending ----
  for (int k = 2; k <= SORTN; k <<= 1) {
    for (int j = k >> 1; j > 0; j >>= 1) {
      for (int p = 0; p < SORTN / NMS_THREADS; ++p) {
        int i = p * NMS_THREADS + t;
        int ixj = i ^ j;
        if (ixj > i) {
          unsigned long long a = sh_sort[i], b = sh_sort[ixj];
          bool desc = ((i & k) == 0);
          if (desc ? (a < b) : (a > b)) { sh_sort[i] = b; sh_sort[ixj] = a; }
        }
      }
      __syncthreads();
    }
  }

  // ---- materialize top-300 candidates (re-decode their boxes) ----
  for (int m = t; m < MAX_CAND; m += NMS_THREADS) {
    unsigned long long e = sh_sort[m];
    if (e != 0ull) {
      int idx = (int)(0xFFFFFFFFu - (uint32_t)(e & 0xFFFFFFFFull));
      float score, box[4];
      decode_one(idx, sm, bp, rois, w_img, h_img, score, box);
      sh_idx[m] = idx;
      sh_box[m][0] = box[0]; sh_box[m][1] = box[1];
      sh_box[m][2] = box[2]; sh_box[m][3] = box[3];
      sh_keep[m] = 1;
    } else {
      sh_idx[m] = -1; sh_keep[m] = 0;
      sh_box[m][0] = sh_box[m][1] = sh_box[m][2] = sh_box[m][3] = 0.0f;
    }
  }
  __syncthreads();

  // ---- parallel adjacency: adj[i] bit j set iff j>i and IoU(i,j) > OVERLAP ----
  for (int m = t; m < MAX_CAND; m += NMS_THREADS) {
    uint32_t row[KEEPW];
#pragma unroll
    for (int w = 0; w < KEEPW; ++w) row[w] = 0u;
    if (sh_keep[m]) {
      float ax0 = sh_box[m][0], ay0 = sh_box[m][1];
      float ax1 = sh_box[m][2], ay1 = sh_box[m][3];
      float areaA = (ax1 - ax0) * (ay1 - ay0);
      for (int j = m + 1; j < MAX_CAND; ++j) {
        float bx0 = sh_box[j][0], by0 = sh_box[j][1];
        float bx1 = sh_box[j][2], by1 = sh_box[j][3];
        float iw = fmaxf(fminf(ax1, bx1) - fmaxf(ax0, bx0), 0.0f);
        float ih = fmaxf(fminf(ay1, by1) - fmaxf(ay0, by0), 0.0f);
        float inter = iw * ih;
        float areaB = (bx1 - bx0) * (by1 - by0);
        float iou = inter / (areaA + areaB - inter);
        if (iou > OVERLAP) row[j >> 5] |= (1u << (j & 31));
      }
    }
#pragma unroll
    for (int w = 0; w < KEEPW; ++w) sh_adj[m][w] = row[w];
  }
  __syncthreads();

  // ---- exact greedy recurrence on bitmasks, single thread, no barriers ----
  if (t == 0) {
    uint32_t keep[KEEPW];
#pragma unroll
    for (int w = 0; w < KEEPW; ++w) keep[w] = 0u;
    for (int m = 0; m < MAX_CAND; ++m)
      if (sh_keep[m]) keep[m >> 5] |= (1u << (m & 31));
    for (int i = 0; i < MAX_CAND; ++i) {
      if ((keep[i >> 5] >> (i & 31)) & 1u) {
#pragma unroll
        for (int w = 0; w < KEEPW; ++w) keep[w] &= ~sh_adj[i][w];
        keep[i >> 5] |= (1u << (i & 31));   // adj rows have only j>i bits, but be explicit
      }
    }
    int cnt = 0;
    for (int m = 0; m < MAX_CAND; ++m) {
      int k = (keep[m >> 5] >> (m & 31)) & 1;
      sh_keep[m] = k;
      sh_ord[m]  = k ? cnt : TOP_N;
      cnt += k;
    }
  }
  __syncthreads();

  // ---- scatter output rows [0, box(4), probs(5)] ----
  for (int m = t; m < MAX_CAND; m += NMS_THREADS) {
    if (sh_keep[m] && sh_ord[m] < TOP_N) {
      float* row = out + (size_t)sh_ord[m] * 10;
      row[0] = 0.0f;
      row[1] = sh_box[m][0]; row[2] = sh_box[m][1];
      row[3] = sh_box[m][2]; row[4] = sh_box[m][3];
      const float* s = sm + (size_t)sh_idx[m] * NCLS;
#pragma unroll
      for (int c = 0; c < NCLS; ++c) row[5 + c] = s[c];
    }
  }
}

extern "C" void kernel_launch(void* const* d_in, const int* in_sizes, int n_in,
                              void* d_out, int out_size, void* d_ws, size_t ws_size,
                              hipStream_t stream) {
  (void)n_in; (void)out_size; (void)ws_size;
  const float* sm      = (const float*)d_in[0];   // (N, 5) softmax
  const float* bp      = (const float*)d_in[1];   // (N, 20) bbox deltas
  const float* rois    = (const float*)d_in[2];   // (N, 5)
  const float* im_info = (const float*)d_in[3];   // (2,)
  float*    out = (float*)d_out;                  // (100, 10)
  uint32_t* ws  = (uint32_t*)d_ws;                // ~295 KB + 4 MB keys

  int n = in_sizes[0] / NCLS;
  int nb = (n + 255) / 256;
  int zb = (WS_ZERO_U32 + 255) / 256;

  k_zero    <<<zb, 256, 0, stream>>>(ws, out);
  k_hist    <<<nb, 256, 0, stream>>>(sm, bp, rois, im_info, ws, n);
  k_select  <<<1, 1024, 0, stream>>>(ws);
  k_compact <<<nb, 256, 0, stream>>>(ws, n);
  k_sort_nms<<<1, NMS_THREADS, 0, stream>>>(sm, bp, rois, im_info, ws, out);
}